// S6Block_78623671320856
// MI455X (gfx1250) — compile-verified
//
#include <hip/hip_runtime.h>
#include <hip/hip_bf16.h>

// ---- problem constants (fixed by reference) ----
#define BB   2
#define CC   96
#define HH   64
#define WWW  64
#define LL   4096
#define NN   16
#define RR   6
#define KK   4
#define DBCROWS 38          // R + 2N
#define MPAD 48             // 38 padded to 3 WMMA M-tiles
#define LT   64             // L tile per block in projection GEMM
#define CH   4              // scan chunks
#define CHL  (LL / CH)      // 1024

typedef float v2f __attribute__((ext_vector_type(2)));
typedef float v8f __attribute__((ext_vector_type(8)));

// direction index map: position in x / output for scan position l of direction k
__device__ __forceinline__ int tr_idx(int l) { return ((l & 63) << 6) | (l >> 6); }
__device__ __forceinline__ int dir_idx(int k, int l) {
    switch (k) {
        case 0:  return l;
        case 1:  return tr_idx(l);
        case 2:  return LL - 1 - l;
        default: return tr_idx(LL - 1 - l);
    }
}

// ---------------------------------------------------------------------------
// Kernel 1: per (b,k,Ltile): DBC = W(38x96) @ Xs(96x64) via v_wmma_f32_16x16x4,
// then delta = softplus(Wd(96x6) @ deltas(6x64) + bias) -> (B,K,C,L),
// and B/C rows transposed out as (B,K,L,N).
// ---------------------------------------------------------------------------
__global__ __launch_bounds__(256)
void s6_proj_kernel(const float* __restrict__ x,
                    const float* __restrict__ wproj,   // (K,38,96)
                    const float* __restrict__ wdelta,  // (K,96,6)
                    const float* __restrict__ dbias,   // (K,96)
                    float* __restrict__ delta_out,     // (B,K,C,L)
                    float* __restrict__ Bt,            // (B,K,L,N)
                    float* __restrict__ Ct)            // (B,K,L,N)
{
    __shared__ float sW[MPAD][CC];        // 18.0 KB, zero-padded rows 38..47
    __shared__ float sX[CC][LT + 1];      // 24.4 KB
    __shared__ float sDBC[MPAD][LT + 1];  // 12.2 KB
    __shared__ float sWd[CC][RR];         //  2.3 KB

    const int tid = threadIdx.x;
    const int l0  = blockIdx.x * LT;
    const int k   = blockIdx.y;
    const int b   = blockIdx.z;

    // stage weights
    for (int i = tid; i < MPAD * CC; i += 256) {
        int m = i / CC, c = i % CC;
        sW[m][c] = (m < DBCROWS) ? wproj[(k * DBCROWS + m) * CC + c] : 0.0f;
    }
    for (int i = tid; i < CC * RR; i += 256) {
        int c = i / RR, r = i % RR;
        sWd[c][r] = wdelta[(k * CC + c) * RR + r];
    }
    // stage x tile with direction mapping (xs built on the fly)
    for (int i = tid; i < CC * LT; i += 256) {
        int c = i / LT, col = i % LT;
        sX[c][col] = x[((size_t)b * CC + c) * LL + dir_idx(k, l0 + col)];
    }
    __syncthreads();

    // GEMM: 3 M-tiles x 4 N-tiles over 8 waves, K=96 in steps of 4
    const int wave = tid >> 5;
    const int lane = tid & 31;
    const int lm   = lane & 15;
    const int lh   = lane >> 4;   // half-wave select
    for (int t = wave; t < (MPAD / 16) * (LT / 16); t += 8) {
        const int mt = t / (LT / 16);
        const int nt = t % (LT / 16);
        v8f acc = {};
        for (int kk = 0; kk < CC; kk += 4) {
            v2f a, bf;
            // A 16x4 f32: lanes 0-15 = M rows; VGPR j holds K = 2*lh + j
            a.x  = sW[mt * 16 + lm][kk + 2 * lh + 0];
            a.y  = sW[mt * 16 + lm][kk + 2 * lh + 1];
            // B 4x16 f32 (mirror layout): lane = N col; VGPR j holds K = 2*lh + j
            bf.x = sX[kk + 2 * lh + 0][nt * 16 + lm];
            bf.y = sX[kk + 2 * lh + 1][nt * 16 + lm];
            acc = __builtin_amdgcn_wmma_f32_16x16x4_f32(
                      false, a, false, bf, (short)0, acc, false, false);
        }
        // D 16x16 f32: VGPR i -> row M = i + 8*lh, lane -> col
        #pragma unroll
        for (int i = 0; i < 8; ++i)
            sDBC[mt * 16 + i + 8 * lh][nt * 16 + lm] = acc[i];
    }
    __syncthreads();

    // B/C rows (6..21, 22..37) -> transposed (B,K,L,N)
    for (int i = tid; i < LT * NN; i += 256) {
        int col = i / NN, n = i % NN;
        size_t base = (((size_t)(b * KK + k) * LL + (l0 + col)) * NN + n);
        Bt[base] = sDBC[RR + n][col];
        Ct[base] = sDBC[RR + NN + n][col];
    }
    // delta = softplus(Wd @ deltas + bias): K-dim is only 6 -> VALU dot
    for (int i = tid; i < CC * LT; i += 256) {
        int c = i / LT, col = i % LT;
        float s = 0.0f;
        #pragma unroll
        for (int r = 0; r < RR; ++r) s += sWd[c][r] * sDBC[r][col];
        s += dbias[k * CC + c];
        float sp = (s > 20.0f) ? s : log1pf(__expf(s));
        delta_out[(((size_t)(b * KK + k) * CC + c) * LL) + l0 + col] = sp;
    }
}

// ---------------------------------------------------------------------------
// Kernel 2: per-chunk scan carries. Half-wave = one (b,k,c,chunk); lane = state n.
// carryA = prod(exp(delta*A_n)) over chunk, carryH = final local state.
// ---------------------------------------------------------------------------
__global__ __launch_bounds__(256)
void s6_scan_carry_kernel(const float* __restrict__ x,
                          const float* __restrict__ delta,
                          const float* __restrict__ Bt,
                          const float* __restrict__ A_log,
                          float* __restrict__ carryA,
                          float* __restrict__ carryH)
{
    const int tid  = threadIdx.x;
    const int unit = (blockIdx.x * 256 + tid) >> 4;   // (ch, chunk)
    const int n    = tid & 15;
    if (unit >= BB * KK * CC * CH) return;
    const int chunk = unit % CH;
    const int ch    = unit / CH;          // (b*K + k)*C + c
    const int c     = ch % CC;
    const int bk    = ch / CC;
    const int k     = bk % KK;
    const int b     = bk / KK;
    const int d     = k * CC + c;

    const float An = -__expf(A_log[d * NN + n]);
    const float* dptr = delta + ((size_t)bk * CC + c) * LL;
    const float* bptr = Bt + (size_t)bk * LL * NN;
    const float* xptr = x + ((size_t)b * CC + c) * LL;

    float h = 0.0f, ap = 1.0f;
    const int lbeg = chunk * CHL, lend = lbeg + CHL;
    for (int l = lbeg; l < lend; ++l) {
        __builtin_prefetch(dptr + ((l + 64) & (LL - 1)), 0, 0);
        __builtin_prefetch(bptr + (size_t)((l + 16) & (LL - 1)) * NN, 0, 0);
        float ld = dptr[l];
        float u  = xptr[dir_idx(k, l)];
        float Bn = bptr[(size_t)l * NN + n];
        float dA = __expf(ld * An);
        h  = dA * h + (ld * u) * Bn;
        ap *= dA;
    }
    size_t cidx = (size_t)unit * NN + n;
    carryA[cidx] = ap;
    carryH[cidx] = h;
}

// ---------------------------------------------------------------------------
// Kernel 3: chain carries across chunks -> per-chunk initial states.
// ---------------------------------------------------------------------------
__global__ __launch_bounds__(256)
void s6_combine_kernel(const float* __restrict__ carryA,
                       const float* __restrict__ carryH,
                       float* __restrict__ initH)
{
    int i = blockIdx.x * 256 + threadIdx.x;     // (ch, n)
    if (i >= BB * KK * CC * NN) return;
    int ch = i / NN, n = i % NN;
    float Hp = 0.0f;
    for (int j = 0; j < CH; ++j) {
        size_t idx = (size_t)(ch * CH + j) * NN + n;
        initH[idx] = Hp;
        Hp = carryA[idx] * Hp + carryH[idx];
    }
}

// ---------------------------------------------------------------------------
// Kernel 4: full scan with correct initial state, fused y = sum_n h*C + D*u,
// scattered directly into the direction-inverse output positions.
// ---------------------------------------------------------------------------
__global__ __launch_bounds__(256)
void s6_scan_out_kernel(const float* __restrict__ x,
                        const float* __restrict__ delta,
                        const float* __restrict__ Bt,
                        const float* __restrict__ Ct,
                        const float* __restrict__ A_log,
                        const float* __restrict__ Ds,
                        const float* __restrict__ initH,
                        float* __restrict__ out)
{
    const int tid  = threadIdx.x;
    const int unit = (blockIdx.x * 256 + tid) >> 4;
    const int n    = tid & 15;
    if (unit >= BB * KK * CC * CH) return;
    const int chunk = unit % CH;
    const int ch    = unit / CH;
    const int c     = ch % CC;
    const int bk    = ch / CC;
    const int k     = bk % KK;
    const int b     = bk / KK;
    const int d     = k * CC + c;

    const float An = -__expf(A_log[d * NN + n]);
    const float Dd = Ds[d];
    const float* dptr = delta + ((size_t)bk * CC + c) * LL;
    const float* bptr = Bt + (size_t)bk * LL * NN;
    const float* cptr = Ct + (size_t)bk * LL * NN;
    const float* xptr = x + ((size_t)b * CC + c) * LL;
    // output slot: k=0->out0, k=1->out2 (transposed), k=2->out1 (flipped), k=3->out3
    const int slot = (k == 0) ? 0 : (k == 1) ? 2 : (k == 2) ? 1 : 3;
    float* optr = out + ((size_t)(slot * BB + b) * CC + c) * LL;

    float h = initH[(size_t)unit * NN + n];
    const int lbeg = chunk * CHL, lend = lbeg + CHL;
    for (int l = lbeg; l < lend; ++l) {
        __builtin_prefetch(dptr + ((l + 64) & (LL - 1)), 0, 0);
        __builtin_prefetch(cptr + (size_t)((l + 16) & (LL - 1)) * NN, 0, 0);
        float ld  = dptr[l];
        int   idx = dir_idx(k, l);           // same map for u read and y write
        float u   = xptr[idx];
        float Bn  = bptr[(size_t)l * NN + n];
        float Cn  = cptr[(size_t)l * NN + n];
        float dA  = __expf(ld * An);
        h = dA * h + (ld * u) * Bn;
        float v = h * Cn;
        v += __shfl_xor(v, 1);
        v += __shfl_xor(v, 2);
        v += __shfl_xor(v, 4);
        v += __shfl_xor(v, 8);
        if (n == 0) optr[idx] = v + Dd * u;
    }
}

// ---------------------------------------------------------------------------
extern "C" void kernel_launch(void* const* d_in, const int* in_sizes, int n_in,
                              void* d_out, int out_size, void* d_ws, size_t ws_size,
                              hipStream_t stream) {
    const float* x      = (const float*)d_in[0];
    const float* wproj  = (const float*)d_in[1];
    const float* wdelta = (const float*)d_in[2];
    const float* dbias  = (const float*)d_in[3];
    const float* A_log  = (const float*)d_in[4];
    const float* Ds     = (const float*)d_in[5];
    float* out = (float*)d_out;

    float* ws     = (float*)d_ws;
    float* delta  = ws;                                      // B*K*C*L
    float* Bt     = delta  + (size_t)BB * KK * CC * LL;      // B*K*L*N
    float* Ct     = Bt     + (size_t)BB * KK * LL * NN;      // B*K*L*N
    float* carryA = Ct     + (size_t)BB * KK * LL * NN;      // B*K*C*CH*N
    float* carryH = carryA + (size_t)BB * KK * CC * CH * NN;
    float* initH  = carryH + (size_t)BB * KK * CC * CH * NN;

    dim3 gProj(LL / LT, KK, BB);
    s6_proj_kernel<<<gProj, 256, 0, stream>>>(x, wproj, wdelta, dbias, delta, Bt, Ct);

    const int units = BB * KK * CC * CH;                     // 3072 half-wave units
    s6_scan_carry_kernel<<<units * 16 / 256, 256, 0, stream>>>(x, delta, Bt, A_log,
                                                               carryA, carryH);
    s6_combine_kernel<<<(BB * KK * CC * NN + 255) / 256, 256, 0, stream>>>(carryA, carryH,
                                                                           initH);
    s6_scan_out_kernel<<<units * 16 / 256, 256, 0, stream>>>(x, delta, Bt, Ct, A_log, Ds,
                                                             initH, out);
}